// MyModel_61933428411833
// MI455X (gfx1250) — compile-verified
//
#include <hip/hip_runtime.h>
#include <stdint.h>

#define HW      96
#define CH      320
#define PLANE   (HW * HW)          // 9216 floats per (b,c) plane
#define NPLANES (16 * CH)          // 5120 planes
#define TPB     256                // 8 waves (wave32)

// Padded LDS tile: rows -2..97 (100 rows), stride 104 floats (416 B,
// multiple of 16 B so async B128 interior writes stay aligned).
#define TSTRIDE 104
#define TROWS   100
#define TILEN   (TROWS * TSTRIDE)  // 10400 floats = 41.6 KB

typedef float v2f __attribute__((ext_vector_type(2)));
typedef float v8f __attribute__((ext_vector_type(8)));

template <typename T>
__device__ __forceinline__ T splatf(float s) { T r = {}; r += s; return r; }

// ---------------------------------------------------------------------------
// Reduce two fp32 accumulators across a 256-thread block and write 2 floats.
// Final 32-lane stage is one V_WMMA_F32_16X16X4_F32:
//   A vgpr0 = acc0, A vgpr1 = acc1 (exact 16x4 f32 A layout);
//   B is a 0/1 selector: K0/K2 -> column 0, K1/K3 -> column 1, so
//   C[m,0] = acc0[m]+acc0[m+16], C[m,1] = acc1[m]+acc1[m+16].
//   8 in-lane adds + lane0+lane16 / lane1+lane17 finish the totals.
// ---------------------------------------------------------------------------
__device__ __forceinline__ void block_reduce_2(float a0, float a1,
                                               float* red /* shared 2*TPB */,
                                               float* out /* 2 floats */) {
  const int t = threadIdx.x;
  red[t]       = a0;
  red[TPB + t] = a1;
  __syncthreads();
  if (t < 32) {
    float s0 = 0.f, s1 = 0.f;
#pragma unroll
    for (int i = 0; i < TPB / 32; ++i) {
      s0 += red[t + 32 * i];
      s1 += red[TPB + t + 32 * i];
    }
#if __has_builtin(__builtin_amdgcn_wmma_f32_16x16x4_f32)
    v2f A; A.x = s0; A.y = s1;
    v2f B;
    B.x = (t == 0 || t == 16) ? 1.f : 0.f;
    B.y = (t == 1 || t == 17) ? 1.f : 0.f;
    v8f C = {};
    v8f D = __builtin_amdgcn_wmma_f32_16x16x4_f32(
        false, A, false, B, (short)0, C, false, false);
    float s = D[0] + D[1] + D[2] + D[3] + D[4] + D[5] + D[6] + D[7];
    red[t] = s;                              // LDS is in-order within a wave
    if (t == 0) out[0] = red[0] + red[16];
    if (t == 1) out[1] = red[1] + red[17];
#else
    for (int off = 16; off > 0; off >>= 1) {
      s0 += __shfl_down(s0, off);
      s1 += __shfl_down(s1, off);
    }
    if (t == 0) { out[0] = s0; out[1] = s1; }
#endif
  }
}

// ---------------------------------------------------------------------------
// NITER-row sliding-window 5x5 conv strip + diff accumulation; loads are
// UNCONDITIONAL (halo-padded tile). T = float2 -> packed fp32 math.
// ld(r, v) returns element(s) of the window row r, tap v.
// ---------------------------------------------------------------------------
template <typename T, int NITER, typename LD>
__device__ __forceinline__ void conv_strip(const LD& ld,
                                           const float* __restrict__ w,
                                           int r0, T& acc0, T& acc1) {
  T win[5][5];
#pragma unroll
  for (int u = 0; u < 5; ++u)
#pragma unroll
    for (int v = 0; v < 5; ++v)
      win[u][v] = ld(r0 - 2 + u, v);

#pragma unroll
  for (int i = 0; i < NITER; ++i) {
    T a = {};
#pragma unroll
    for (int u = 0; u < 5; ++u)
#pragma unroll
      for (int v = 0; v < 5; ++v)
        a = __builtin_elementwise_fma(win[u][v], splatf<T>(w[5 * u + v]), a);

    T xc = win[2][2];
    T z  = {};
    T y1 = __builtin_elementwise_max(xc, a);
    T y2 = xc + __builtin_elementwise_max(a - xc, z);   // x + relu(a-x)
    T y3 = a  + __builtin_elementwise_max(xc - a, z);   // a + relu(x-a)
    acc0 += y1 - y2;
    acc1 += y1 - y3;

    if (i < NITER - 1) {             // last slide would read a dead row
#pragma unroll
      for (int u = 0; u < 4; ++u)
#pragma unroll
        for (int v = 0; v < 5; ++v)
          win[u][v] = win[u + 1][v];
#pragma unroll
      for (int v = 0; v < 5; ++v)
        win[4][v] = ld(r0 + i + 3, v);
    }
  }
}

// ---------------------------------------------------------------------------
// Kernel 1: one workgroup per (batch, channel) plane.
// Zero the halo (296 float4 slots), async-copy the 96x96 interior into the
// padded tile (9 B128 chunks/thread), then run the branch-free fully-packed
// conv: cols (tx, tx+32) as a column pair, col tx+64 as a row-split pair.
// ---------------------------------------------------------------------------
__global__ __launch_bounds__(TPB) void dwconv_diff_kernel(
    const float* __restrict__ x, const float* __restrict__ W,
    float* __restrict__ ws) {
  __shared__ __attribute__((aligned(16))) float tile[TILEN];
  __shared__ float red[2 * TPB];

  const int t = threadIdx.x;
  const int p = blockIdx.x;          // plane index = b*CH + c
  const int c = p % CH;

  // ---- zero halo: full rows {0,1,98,99} (26 float4 each) + 2 side slots
  //      (col4 0 and 25) for the 96 interior rows = 104 + 192 = 296 slots.
  {
    float4* t4 = (float4*)tile;
    const float4 z4 = {0.f, 0.f, 0.f, 0.f};
#pragma unroll
    for (int i = t; i < 296; i += TPB) {
      int slot;
      if (i < 104) {
        const int fr = i / 26;                       // 0..3
        const int row = (fr < 2) ? fr : (96 + fr);   // 0,1,98,99
        slot = row * (TSTRIDE / 4) + (i % 26);
      } else {
        const int j   = i - 104;
        const int row = 2 + (j >> 1);                // 2..97
        slot = row * (TSTRIDE / 4) + ((j & 1) ? 25 : 0);
      }
      t4[slot] = z4;
    }
  }

  // ---- async interior copy: 96 rows x 24 B128 chunks = 2304 = 9/thread ----
  const char*    src  = (const char*)(x + (size_t)p * PLANE);
  const uint32_t lds0 = (uint32_t)(uintptr_t)tile;   // low 32 bits = LDS addr
#pragma unroll
  for (int i = 0; i < 9; ++i) {
    const int id  = i * TPB + t;
    const int row = id / 24;
    const int k   = id % 24;
    const uint32_t loff = (uint32_t)((row + 2) * (TSTRIDE * 4) + 16 + k * 16);
    const void*    g    = (const void*)(src + row * (HW * 4) + k * 16);
    asm volatile("global_load_async_to_lds_b128 %0, %1, off"
                 :: "v"(lds0 + loff), "v"(g) : "memory");
  }
#if __has_builtin(__builtin_amdgcn_s_wait_asynccnt)
  __builtin_amdgcn_s_wait_asynccnt(0);
#else
  asm volatile("s_wait_asynccnt 0" ::: "memory");
#endif
  __syncthreads();

  // 25 weights for this channel (uniform -> scalar loads)
  float w[25];
#pragma unroll
  for (int k = 0; k < 25; ++k) w[k] = W[c * 25 + k];

  const int tx = t & 31;             // lane -> column within group
  const int r0 = (t >> 5) * 12;      // wave -> 12 consecutive output rows

  // padded address of (x row r, x col cc, tap v) = (r+2)*TSTRIDE + cc + 2 + v
  v2f acc0 = {}, acc1 = {};

  // strip A: columns tx and tx+32 paired, 12 output rows
  {
    auto ld2 = [&](int r, int v) -> v2f {
      const float* rowp = tile + (r + 2) * TSTRIDE + 2 + tx + v;
      v2f o;
      o.x = rowp[0];
      o.y = rowp[32];
      return o;
    };
    conv_strip<v2f, 12>(ld2, w, r0, acc0, acc1);
  }

  // strip B: column tx+64, rows r0..r0+5 (.x) paired with r0+6..r0+11 (.y)
  {
    auto ld2s = [&](int r, int v) -> v2f {
      const float* colp = tile + 2 + tx + 64 + v;
      v2f o;
      o.x = colp[(r + 2) * TSTRIDE];
      o.y = colp[(r + 8) * TSTRIDE];   // same window, 6 rows below
      return o;
    };
    conv_strip<v2f, 6>(ld2s, w, r0, acc0, acc1);
  }

  block_reduce_2(acc0.x + acc0.y, acc1.x + acc1.y, red, ws + 2 * (size_t)p);
}

// ---------------------------------------------------------------------------
// Kernel 2: deterministic final reduction 5120 x 2 partials -> d_out[0..1].
// ---------------------------------------------------------------------------
__global__ __launch_bounds__(TPB) void final_reduce_kernel(
    const float* __restrict__ ws, float* __restrict__ out) {
  __shared__ float red[2 * TPB];
  const int t = threadIdx.x;
  const v2f* ws2 = (const v2f*)ws;   // {diff01, diff02} pairs, b64 loads
  v2f a = {};
  for (int i = t; i < NPLANES; i += TPB) a += ws2[i];
  block_reduce_2(a.x, a.y, red, out);
}

extern "C" void kernel_launch(void* const* d_in, const int* in_sizes, int n_in,
                              void* d_out, int out_size, void* d_ws,
                              size_t ws_size, hipStream_t stream) {
  (void)in_sizes; (void)n_in; (void)out_size; (void)ws_size;
  const float* x = (const float*)d_in[0];
  const float* W = (const float*)d_in[1];
  float* ws  = (float*)d_ws;   // needs 5120*2*4 = 40 KB scratch
  float* out = (float*)d_out;  // 2 fp32 scalars

  dwconv_diff_kernel<<<NPLANES, TPB, 0, stream>>>(x, W, ws);
  final_reduce_kernel<<<1, TPB, 0, stream>>>(ws, out);
}